// Actor_78228534329752
// MI455X (gfx1250) — compile-verified
//
#include <hip/hip_runtime.h>
#include <math.h>

// CDNA5 / gfx1250: wave32, WMMA f32 16x16x4.
typedef __attribute__((ext_vector_type(2))) float v2f;
typedef __attribute__((ext_vector_type(8))) float v8f;

#define HD 64   // hidden dim
#define FI 6    // input feature dim

// ---------------- degree / normalization ----------------
__global__ void k_init_deg(float* deg, int n) {
    int i = blockIdx.x * blockDim.x + threadIdx.x;
    if (i < n) deg[i] = 1.0f;                 // self-loop weight 1 at target
}

__global__ void k_deg_accum(const int* __restrict__ col, const float* __restrict__ w,
                            float* deg, int e) {
    int i = blockIdx.x * blockDim.x + threadIdx.x;
    if (i < e) atomicAdd(&deg[col[i]], w[i]);
}

__global__ void k_dis(float* d, int n) {     // in place: deg -> D^{-1/2}
    int i = blockIdx.x * blockDim.x + threadIdx.x;
    if (i < n) {
        float v = d[i];
        d[i] = (v > 0.0f) ? rsqrtf(fmaxf(v, 1e-30f)) : 0.0f;
    }
}

// ---------------- layer-1 GEMM (K=6, scalar) ----------------
__global__ void k_gemm_f6(const float* __restrict__ x, const float* __restrict__ W,
                          float* __restrict__ xw, int n) {
    int t = blockIdx.x * blockDim.x + threadIdx.x;
    if (t >= n * HD) return;
    int i = t >> 6, j = t & 63;
    const float* xr = x + i * FI;
    float s = 0.0f;
#pragma unroll
    for (int k = 0; k < FI; ++k) s += xr[k] * W[k * HD + j];
    xw[t] = s;
}

// ---------------- 64x64 GEMM via V_WMMA_F32_16X16X4_F32 ----------------
// A:[nrows,64] row-major, B:[64,64] row-major, C:[nrows,64].
// One 16x16 output tile per wave; 16 chained K=4 WMMA ops.
__global__ void k_gemm_wmma(const float* __restrict__ A, const float* __restrict__ Bm,
                            float* __restrict__ C, int ntiles) {
    int wid = (int)((blockIdx.x * blockDim.x + threadIdx.x) >> 5);
    if (wid >= ntiles) return;                // wave-uniform: EXEC stays all-ones
    int lane = threadIdx.x & 31;
    int half = lane >> 4;                     // 0: K pair (k,k+1), 1: (k+2,k+3)
    int l    = lane & 15;
    int tm = wid >> 2;                        // 64/16 = 4 column tiles
    int tn = wid & 3;
    int m0 = tm << 4, n0 = tn << 4;

    const float* arow = A + (size_t)(m0 + l) * HD;   // A: M = l, per ISA 16x4 layout
    const float* bcol = Bm + n0 + l;                  // B: N = l
    v8f c = {};
#pragma unroll
    for (int k = 0; k < HD; k += 4) {
        int ka = k + 2 * half;
        v2f a, b;
        a.x = arow[ka];                 // A[M][ka]
        a.y = arow[ka + 1];             // A[M][ka+1]
        b.x = bcol[(size_t)ka * HD];    // B[ka][N]
        b.y = bcol[(size_t)(ka + 1) * HD];
        c = __builtin_amdgcn_wmma_f32_16x16x4_f32(
                /*neg_a=*/false, a, /*neg_b=*/false, b,
                /*c_mod=*/(short)0, c, /*reuse_a=*/false, /*reuse_b=*/false);
    }
    // D layout: VGPR j -> row m0 + half*8 + j, col n0 + l
    float* crow = C + (size_t)(m0 + half * 8) * HD + n0 + l;
#pragma unroll
    for (int j = 0; j < 8; ++j) crow[(size_t)j * HD] = c[j];
}

// ---------------- aggregation ----------------
__global__ void k_agg_init(const float* __restrict__ xw, const float* __restrict__ dis,
                           const float* __restrict__ b, float* __restrict__ agg, int n) {
    int t = blockIdx.x * blockDim.x + threadIdx.x;
    if (t >= n * HD) return;
    int i = t >> 6, j = t & 63;
    float d = dis[i];
    agg[t] = d * d * xw[t] + b[j];            // self-loop term + bias
}

// 16 threads per edge, 4 features each (float4 gather, 4 f32 atomics).
__global__ void k_scatter(const int* __restrict__ row, const int* __restrict__ col,
                          const float* __restrict__ w, const float* __restrict__ dis,
                          const float* __restrict__ xw, float* __restrict__ agg, int e) {
    int t = blockIdx.x * blockDim.x + threadIdx.x;
    if (t >= e * 16) return;
    int eid = t >> 4;
    int f0  = (t & 15) << 2;
    int r = row[eid], cdst = col[eid];
    float coef = dis[r] * w[eid] * dis[cdst];
    const float4 v = *(const float4*)(xw + (size_t)r * HD + f0);
    float* ap = agg + (size_t)cdst * HD + f0;
    atomicAdd(ap + 0, coef * v.x);
    atomicAdd(ap + 1, coef * v.y);
    atomicAdd(ap + 2, coef * v.z);
    atomicAdd(ap + 3, coef * v.w);
}

__global__ void k_sigmoid(const float* __restrict__ agg, float* __restrict__ h, int n64) {
    int t = blockIdx.x * blockDim.x + threadIdx.x;
    if (t < n64) h[t] = 1.0f / (1.0f + expf(-agg[t]));
}

// ---------------- head: logits + softmax over nodes ----------------
__global__ void k_logits(const float* __restrict__ h, const float* __restrict__ Wl,
                         const float* __restrict__ bl, float* __restrict__ lg, int n) {
    int i = blockIdx.x * blockDim.x + threadIdx.x;
    if (i >= n) return;
    const float* hr = h + (size_t)i * HD;
    float s = bl[0];
#pragma unroll
    for (int f = 0; f < HD; ++f) s += hr[f] * Wl[f];
    lg[i] = s;
}

__global__ void k_reduce_max(const float* __restrict__ lg, float* gmax, int n) {
    __shared__ float s[256];
    float m = -INFINITY;
    for (int i = threadIdx.x; i < n; i += 256) m = fmaxf(m, lg[i]);
    s[threadIdx.x] = m;
    __syncthreads();
    for (int off = 128; off > 0; off >>= 1) {
        if ((int)threadIdx.x < off) s[threadIdx.x] = fmaxf(s[threadIdx.x], s[threadIdx.x + off]);
        __syncthreads();
    }
    if (threadIdx.x == 0) *gmax = s[0];
}

__global__ void k_reduce_sum(const float* __restrict__ lg, const float* __restrict__ gmax,
                             float* gsum, int n) {
    __shared__ float s[256];
    float mx = *gmax;
    float acc = 0.0f;
    for (int i = threadIdx.x; i < n; i += 256) acc += expf(lg[i] - mx);
    s[threadIdx.x] = acc;
    __syncthreads();
    for (int off = 128; off > 0; off >>= 1) {
        if ((int)threadIdx.x < off) s[threadIdx.x] += s[threadIdx.x + off];
        __syncthreads();
    }
    if (threadIdx.x == 0) *gsum = s[0];
}

__global__ void k_softmax_out(const float* __restrict__ lg, const float* __restrict__ gmax,
                              const float* __restrict__ gsum, float* __restrict__ out, int n) {
    int i = blockIdx.x * blockDim.x + threadIdx.x;
    if (i < n) out[i] = expf(lg[i] - *gmax) / *gsum;
}

// ---------------- launch ----------------
static inline unsigned nblk(long long n, int b) { return (unsigned)((n + b - 1) / b); }

extern "C" void kernel_launch(void* const* d_in, const int* in_sizes, int n_in,
                              void* d_out, int out_size, void* d_ws, size_t ws_size,
                              hipStream_t stream) {
    const float* x   = (const float*)d_in[0];
    const int*   eg  = (const int*)d_in[1];
    const float* w   = (const float*)d_in[2];
    const float* W1  = (const float*)d_in[3];
    const float* b1  = (const float*)d_in[4];
    const float* W2  = (const float*)d_in[5];
    const float* b2  = (const float*)d_in[6];
    const float* W3  = (const float*)d_in[7];
    const float* b3  = (const float*)d_in[8];
    const float* Wl  = (const float*)d_in[9];
    const float* bl  = (const float*)d_in[10];
    float* out = (float*)d_out;

    const int E = in_sizes[2];
    const int N = in_sizes[0] / FI;
    const int* row = eg;          // edges[0,:]
    const int* col = eg + E;      // edges[1,:]

    float* ws   = (float*)d_ws;
    float* dis  = ws;                          // N   (deg -> D^-1/2)
    float* xw   = dis  + (size_t)N;            // N*64
    float* agg  = xw   + (size_t)N * HD;       // N*64
    float* hbuf = agg  + (size_t)N * HD;       // N*64
    float* lg   = hbuf + (size_t)N * HD;       // N
    float* gmax = lg + N;                      // 1
    float* gsum = gmax + 1;                    // 1

    const int B = 256;
    const int ntiles = ((N + 15) / 16) * (HD / 16);   // 16x16 output tiles

    // normalization
    k_init_deg <<<nblk(N, B), B, 0, stream>>>(dis, N);
    k_deg_accum<<<nblk(E, B), B, 0, stream>>>(col, w, dis, E);
    k_dis      <<<nblk(N, B), B, 0, stream>>>(dis, N);

    // ---- layer 1: x[N,6] @ W1[6,64] ----
    k_gemm_f6  <<<nblk((long long)N * HD, B), B, 0, stream>>>(x, W1, xw, N);
    k_agg_init <<<nblk((long long)N * HD, B), B, 0, stream>>>(xw, dis, b1, agg, N);
    k_scatter  <<<nblk((long long)E * 16, B), B, 0, stream>>>(row, col, w, dis, xw, agg, E);
    k_sigmoid  <<<nblk((long long)N * HD, B), B, 0, stream>>>(agg, hbuf, N * HD);

    // ---- layer 2: h[N,64] @ W2[64,64] (WMMA) ----
    k_gemm_wmma<<<nblk((long long)ntiles * 32, B), B, 0, stream>>>(hbuf, W2, xw, ntiles);
    k_agg_init <<<nblk((long long)N * HD, B), B, 0, stream>>>(xw, dis, b2, agg, N);
    k_scatter  <<<nblk((long long)E * 16, B), B, 0, stream>>>(row, col, w, dis, xw, agg, E);
    k_sigmoid  <<<nblk((long long)N * HD, B), B, 0, stream>>>(agg, hbuf, N * HD);

    // ---- layer 3: h[N,64] @ W3[64,64] (WMMA) ----
    k_gemm_wmma<<<nblk((long long)ntiles * 32, B), B, 0, stream>>>(hbuf, W3, xw, ntiles);
    k_agg_init <<<nblk((long long)N * HD, B), B, 0, stream>>>(xw, dis, b3, agg, N);
    k_scatter  <<<nblk((long long)E * 16, B), B, 0, stream>>>(row, col, w, dis, xw, agg, E);
    k_sigmoid  <<<nblk((long long)N * HD, B), B, 0, stream>>>(agg, hbuf, N * HD);

    // ---- head ----
    k_logits     <<<nblk(N, B), B, 0, stream>>>(hbuf, Wl, bl, lg, N);
    k_reduce_max <<<1, 256, 0, stream>>>(lg, gmax, N);
    k_reduce_sum <<<1, 256, 0, stream>>>(lg, gmax, gsum, N);
    k_softmax_out<<<nblk(N, B), B, 0, stream>>>(lg, gmax, gsum, out, N);
}